// MoE_66443144069227
// MI455X (gfx1250) — compile-verified
//
#include <hip/hip_runtime.h>

#define B_   8192
#define IN_  1024
#define OUT_ 256
#define E_   16
#define H_   512
#define K_   3
#define TM_  32          // rows per expert tile (doubles B-fragment reuse)

typedef __attribute__((ext_vector_type(2))) float v2f;
typedef __attribute__((ext_vector_type(8))) float v8f;

// ---------------------------------------------------------------- init
__global__ void moe_init_counts(int* __restrict__ counts) {
    if (threadIdx.x < E_) counts[threadIdx.x] = 0;
}

// ---------------------------------------------------------------- gating logits (fp32, B*E threads)
__global__ __launch_bounds__(256) void moe_gate_logits(
    const float* __restrict__ x, const float* __restrict__ wg,
    float* __restrict__ logits)
{
    int t = blockIdx.x * 256 + threadIdx.x;   // t < B_*E_
    int b = t >> 4;
    int e = t & 15;
    const float* xr = x + (size_t)b * IN_;
    float acc = 0.f;
    for (int i = 0; i < IN_; i += 4) {
        float4 xv = *(const float4*)(xr + i);
        acc = fmaf(xv.x, wg[(i + 0) * E_ + e], acc);
        acc = fmaf(xv.y, wg[(i + 1) * E_ + e], acc);
        acc = fmaf(xv.z, wg[(i + 2) * E_ + e], acc);
        acc = fmaf(xv.w, wg[(i + 3) * E_ + e], acc);
    }
    logits[t] = acc;
}

// ---------------------------------------------------------------- top-3 + softmax + dispatch build
__global__ __launch_bounds__(256) void moe_topk(
    const float* __restrict__ logits,
    float* __restrict__ top_gate, int* __restrict__ top_idx,
    int* __restrict__ counts, int* __restrict__ rows)
{
    int b = blockIdx.x * 256 + threadIdx.x;
    if (b >= B_) return;
    float v[E_];
#pragma unroll
    for (int e = 0; e < E_; ++e) v[e] = logits[b * E_ + e];

    int   idx[K_];
    float val[K_];
    bool  used[E_];
#pragma unroll
    for (int e = 0; e < E_; ++e) used[e] = false;

#pragma unroll
    for (int k = 0; k < K_; ++k) {
        float best = -3.402823466e+38f;
        int   bi   = 0;
#pragma unroll
        for (int e = 0; e < E_; ++e) {
            if (!used[e] && v[e] > best) { best = v[e]; bi = e; }  // ties -> lowest index (jax top_k)
        }
        used[bi] = true;
        idx[k] = bi; val[k] = best;
    }
    float g[K_]; float s = 0.f;
#pragma unroll
    for (int k = 0; k < K_; ++k) { g[k] = expf(val[k] - val[0]); s += g[k]; }
    float inv = 1.f / s;
#pragma unroll
    for (int k = 0; k < K_; ++k) {
        float gk = g[k] * inv;
        top_gate[b * K_ + k] = gk;
        top_idx [b * K_ + k] = idx[k];
        int pos = atomicAdd(&counts[idx[k]], 1);
        rows[(size_t)idx[k] * B_ + pos] = (b << 2) | k;   // pack (row, which-of-3)
    }
}

// ---------------------------------------------------------------- deterministic importance reduction
__global__ __launch_bounds__(256) void moe_importance(
    const float* __restrict__ top_gate, const int* __restrict__ top_idx,
    float* __restrict__ importance)
{
    __shared__ float red[256];
    int e = blockIdx.x;
    float s = 0.f;
    for (int b = threadIdx.x; b < B_; b += 256) {
#pragma unroll
        for (int k = 0; k < K_; ++k)
            if (top_idx[b * K_ + k] == e) s += top_gate[b * K_ + k];
    }
    red[threadIdx.x] = s;
    __syncthreads();
    for (int off = 128; off > 0; off >>= 1) {
        if (threadIdx.x < off) red[threadIdx.x] += red[threadIdx.x + off];
        __syncthreads();
    }
    if (threadIdx.x == 0) importance[e] = red[0];
}

// ---------------------------------------------------------------- cv^2 loss (scalar)
__global__ void moe_loss(const float* __restrict__ importance,
                         const int* __restrict__ counts,
                         float* __restrict__ loss_out)
{
    float mi = 0.f, ml = 0.f;
    for (int e = 0; e < E_; ++e) { mi += importance[e]; ml += (float)counts[e]; }
    mi *= (1.f / E_); ml *= (1.f / E_);
    float vi = 0.f, vl = 0.f;
    for (int e = 0; e < E_; ++e) {
        float di = importance[e] - mi; vi += di * di;
        float dl = (float)counts[e] - ml; vl += dl * dl;
    }
    vi *= (1.f / (E_ - 1)); vl *= (1.f / (E_ - 1));   // ddof=1
    loss_out[0] = vi / (mi * mi + 1e-10f) + vl / (ml * ml + 1e-10f);
}

// ---------------------------------------------------------------- fused per-expert FFN (WMMA f32)
// Block = 256 threads (8 waves). One block = 32 gathered rows of one expert.
// Each W1/W2 B-fragment feeds TWO wmmas (row-halves 0-15 and 16-31): halves
// global load traffic per flop vs a 16-row tile.
// Phase 1: H = relu(X @ W1[e] + b1[e]) -> LDS (wave w owns cols [w*64, w*64+64))
// Phase 2: O = H @ W2[e] + b2[e]       -> gate-scaled scatter to out_buf[(b*3+k)*OUT]
__global__ __launch_bounds__(256) void moe_expert_ffn(
    const float* __restrict__ x,
    const float* __restrict__ W1, const float* __restrict__ b1,
    const float* __restrict__ W2, const float* __restrict__ b2,
    const int* __restrict__ counts, const int* __restrict__ rows,
    const float* __restrict__ top_gate,
    float* __restrict__ out_buf)
{
    const int e   = blockIdx.y;
    const int cnt = counts[e];
    const int n0  = blockIdx.x * TM_;
    if (n0 >= cnt) return;
    int valid = cnt - n0; if (valid > TM_) valid = TM_;

    __shared__ int   rowsS[TM_];
    __shared__ float gateS[TM_];
    __shared__ __align__(16) float Xs[TM_][132];   // 128-wide K chunk, +4 pad (bank spread)
    __shared__ __align__(16) float Hs[TM_][516];   // 512 cols, +4 pad

    if (threadIdx.x < TM_) {
        int r = threadIdx.x;
        int entry = (r < valid) ? rows[(size_t)e * B_ + n0 + r] : -1;
        rowsS[r] = entry;
        gateS[r] = (entry >= 0) ? top_gate[(entry >> 2) * K_ + (entry & 3)] : 0.f;
    }
    __syncthreads();

    const int wv   = threadIdx.x >> 5;   // wave 0..7
    const int lane = threadIdx.x & 31;
    const int m    = lane & 15;          // M index (and N index for B frags)
    const int hi   = lane >> 4;          // K half selector

    // chunk-loader mapping: 8 threads per row, 4 float4 each (coalesced)
    const int lr = threadIdx.x >> 3;     // row 0..31
    const int lc = threadIdx.x & 7;
    int entL = rowsS[lr];
    int bL = ((entL < 0) ? rowsS[0] : entL) >> 2;
    const float4* xrow = (const float4*)(x + (size_t)bL * IN_);

    const float* W1e = W1 + (size_t)e * IN_ * H_;   // [1024][512]
    v8f cacc[2][4];
#pragma unroll
    for (int mh = 0; mh < 2; ++mh)
#pragma unroll
        for (int t = 0; t < 4; ++t) cacc[mh][t] = (v8f){};

    const int col0 = wv * 64 + m;        // phase-1 base column for this lane

    for (int kc = 0; kc < IN_; kc += 128) {
        __syncthreads();                 // previous chunk fully consumed
        const int base = kc >> 2;
#pragma unroll
        for (int j = 0; j < 4; ++j)
            *(float4*)&Xs[lr][lc * 4 + 32 * j] = xrow[base + lc + 8 * j];
        __syncthreads();
#pragma unroll 2
        for (int k = 0; k < 128; k += 4) {
            const int klo = k + 2 * hi;                       // A: vgpr v holds K = klo+v
            v2f a0 = *(const v2f*)&Xs[m][klo];                // rows 0-15   (ds_load_b64)
            v2f a1 = *(const v2f*)&Xs[m + 16][klo];           // rows 16-31
            const float* Wk = W1e + (size_t)(kc + klo) * H_;  // B rows klo, klo+1
#pragma unroll
            for (int t = 0; t < 4; ++t) {
                const float* Wc = Wk + col0 + t * 16;
                v2f bf; bf.x = Wc[0]; bf.y = Wc[H_];
                cacc[0][t] = __builtin_amdgcn_wmma_f32_16x16x4_f32(
                    false, a0, false, bf, (short)0, cacc[0][t], false, false);
                cacc[1][t] = __builtin_amdgcn_wmma_f32_16x16x4_f32(
                    false, a1, false, bf, (short)0, cacc[1][t], false, false);
            }
        }
    }

    // phase-1 epilogue: bias + relu -> Hs
#pragma unroll
    for (int t = 0; t < 4; ++t) {
        int col = wv * 64 + t * 16 + m;
        float bias = b1[e * H_ + col];
#pragma unroll
        for (int mh = 0; mh < 2; ++mh)
#pragma unroll
            for (int r = 0; r < 8; ++r) {
                int mr = r + 8 * hi + 16 * mh;       // C/D: vgpr r -> M = r + 8*hi
                float v = cacc[mh][t][r] + bias;
                Hs[mr][col] = v > 0.f ? v : 0.f;
            }
    }
    __syncthreads();

    // phase 2: O = Hs @ W2[e]
    const float* W2e = W2 + (size_t)e * H_ * OUT_;   // [512][256]
    v8f dacc[2][2];
#pragma unroll
    for (int mh = 0; mh < 2; ++mh)
#pragma unroll
        for (int t = 0; t < 2; ++t) dacc[mh][t] = (v8f){};
    const int col2 = wv * 32 + m;

#pragma unroll 4
    for (int k = 0; k < H_; k += 4) {
        const int klo = k + 2 * hi;
        v2f a0 = *(const v2f*)&Hs[m][klo];
        v2f a1 = *(const v2f*)&Hs[m + 16][klo];
        const float* Wk = W2e + (size_t)klo * OUT_;
#pragma unroll
        for (int t = 0; t < 2; ++t) {
            const float* Wc = Wk + col2 + t * 16;
            v2f bf; bf.x = Wc[0]; bf.y = Wc[OUT_];
            dacc[0][t] = __builtin_amdgcn_wmma_f32_16x16x4_f32(
                false, a0, false, bf, (short)0, dacc[0][t], false, false);
            dacc[1][t] = __builtin_amdgcn_wmma_f32_16x16x4_f32(
                false, a1, false, bf, (short)0, dacc[1][t], false, false);
        }
    }

    // phase-2 epilogue: bias, gate scale, scatter to out_buf[(b*3+k)*OUT + col]
#pragma unroll
    for (int t = 0; t < 2; ++t) {
        int col = wv * 32 + t * 16 + m;
        float bias = b2[e * OUT_ + col];
#pragma unroll
        for (int mh = 0; mh < 2; ++mh)
#pragma unroll
            for (int r = 0; r < 8; ++r) {
                int mr = r + 8 * hi + 16 * mh;
                if (mr < valid) {
                    int entry = rowsS[mr];
                    float g = gateS[mr];
                    size_t oidx = ((size_t)(entry >> 2) * K_ + (entry & 3)) * OUT_ + col;
                    out_buf[oidx] = (dacc[mh][t][r] + bias) * g;
                }
            }
    }
}

// ---------------------------------------------------------------- fixed-order combine
__global__ __launch_bounds__(256) void moe_combine(
    const float* __restrict__ out_buf, float* __restrict__ y)
{
    int t = blockIdx.x * 256 + threadIdx.x;       // t < B_*OUT_
    int o = t & (OUT_ - 1);
    int b = t >> 8;                               // OUT_ == 256
    const float* p = out_buf + (size_t)b * K_ * OUT_ + o;
    y[t] = p[0] + p[OUT_] + p[2 * OUT_];
}

// ---------------------------------------------------------------- launcher
extern "C" void kernel_launch(void* const* d_in, const int* in_sizes, int n_in,
                              void* d_out, int out_size, void* d_ws, size_t ws_size,
                              hipStream_t stream)
{
    (void)in_sizes; (void)n_in; (void)out_size; (void)ws_size;
    const float* x  = (const float*)d_in[0];
    const float* wg = (const float*)d_in[1];
    const float* W1 = (const float*)d_in[2];
    const float* b1 = (const float*)d_in[3];
    const float* W2 = (const float*)d_in[4];
    const float* b2 = (const float*)d_in[5];
    float* y = (float*)d_out;                      // [B*OUT] then loss scalar

    char* ws = (char*)d_ws;
    float* logits     = (float*)ws; ws += (size_t)B_ * E_ * 4;
    int*   top_idx    = (int*)  ws; ws += (size_t)B_ * K_ * 4;
    float* top_gate   = (float*)ws; ws += (size_t)B_ * K_ * 4;
    int*   counts     = (int*)  ws; ws += 64;
    float* importance = (float*)ws; ws += 64;
    int*   rows       = (int*)  ws; ws += (size_t)E_ * B_ * 4;
    float* out_buf    = (float*)ws; ws += (size_t)B_ * K_ * OUT_ * 4;

    moe_init_counts<<<1, 32, 0, stream>>>(counts);
    moe_gate_logits<<<(B_ * E_) / 256, 256, 0, stream>>>(x, wg, logits);
    moe_topk<<<B_ / 256, 256, 0, stream>>>(logits, top_gate, top_idx, counts, rows);
    moe_importance<<<E_, 256, 0, stream>>>(top_gate, top_idx, importance);
    moe_loss<<<1, 1, 0, stream>>>(importance, counts, y + (size_t)B_ * OUT_);
    moe_expert_ffn<<<dim3(B_ / TM_, E_), 256, 0, stream>>>(
        x, W1, b1, W2, b2, counts, rows, top_gate, out_buf);
    moe_combine<<<(B_ * OUT_) / 256, 256, 0, stream>>>(out_buf, y);
}